// HighAgg_13374528160104
// MI455X (gfx1250) — compile-verified
//
#include <hip/hip_runtime.h>
#include <hip/hip_bf16.h>

typedef __attribute__((ext_vector_type(2))) float v2f;
typedef __attribute__((ext_vector_type(8))) float v8f;

// ---------- helpers: order-preserving float<->uint for atomicMax ----------
__device__ __forceinline__ unsigned enc_f(float f) {
    unsigned u = __float_as_uint(f);
    return (u & 0x80000000u) ? ~u : (u | 0x80000000u);
}
__device__ __forceinline__ float dec_f(unsigned u) {
    return (u & 0x80000000u) ? __uint_as_float(u & 0x7FFFFFFFu)
                             : __uint_as_float(~u);
}

// ---------- 1) fold attention vectors into projection weights ----------
// w_src_att[d][h] = sum_k W_src[d, h*64+k] * att_dst[h,k]   (pairs with h_node)
// w_dst_att[d][h] = sum_k W_dst[d, h*64+k] * att_src[h,k]   (pairs with h_high)
__global__ void k_fold_att(const float* __restrict__ Wsrc, const float* __restrict__ Wdst,
                           const float* __restrict__ att_src, const float* __restrict__ att_dst,
                           float* __restrict__ w_src_att, float* __restrict__ w_dst_att) {
    int t = threadIdx.x;          // 256 threads: d = t/4, h = t%4
    int d = t >> 2, h = t & 3;
    float s1 = 0.f, s2 = 0.f;
    #pragma unroll 8
    for (int k = 0; k < 64; ++k) {
        s1 += Wsrc[d * 256 + h * 64 + k] * att_dst[h * 64 + k];
        s2 += Wdst[d * 256 + h * 64 + k] * att_src[h * 64 + k];
    }
    w_src_att[d * 4 + h] = s1;
    w_dst_att[d * 4 + h] = s2;
}

// ---------- 2) scatter-sum features + edge counts ----------
__global__ void k_scatter_mean(const float* __restrict__ F, const int* __restrict__ src,
                               const int* __restrict__ dst, float* __restrict__ sumfeat,
                               float* __restrict__ cnt, int E) {
    int tid = blockIdx.x * blockDim.x + threadIdx.x;
    if (tid >= E * 64) return;
    int e = tid >> 6, d = tid & 63;
    int s = src[e], t = dst[e];
    atomicAdd(&sumfeat[(size_t)t * 64 + d], F[(size_t)s * 64 + d]);
    if (d == 0) atomicAdd(&cnt[t], 1.0f);
}

// ---------- 3) divide by max(count,1) in place ----------
__global__ void k_mean_div(float* __restrict__ sumfeat, const float* __restrict__ cnt, int H) {
    int tid = blockIdx.x * blockDim.x + threadIdx.x;
    if (tid >= H * 64) return;
    sumfeat[tid] = sumfeat[tid] / fmaxf(cnt[tid >> 6], 1.0f);
}

// ---------- 4/10) fp32 WMMA GEMM: C[M,Ncol] = A[M,K] @ B[K,Ncol] (+bias) ----------
// one wave computes one 16x16 output tile with V_WMMA_F32_16X16X4_F32, K stepped by 4
__global__ void k_gemm_wmma_f32(const float* __restrict__ A, const float* __restrict__ B,
                                const float* __restrict__ bias, float* __restrict__ C,
                                int M, int Ncol, int K) {
    int wave = (blockIdx.x * blockDim.x + threadIdx.x) >> 5;
    int lane = threadIdx.x & 31;
    int ntiles = Ncol >> 4;
    int mtiles = (M + 15) >> 4;
    int mt = wave / ntiles, nt = wave % ntiles;
    if (mt >= mtiles) return;               // wave-uniform: EXEC stays all-1s below
    int r  = lane & 15;
    int hi = lane >> 4;                     // K-half select (A: k+2*hi, rows: +8*hi)
    int m = mt * 16 + r; if (m >= M) m = M - 1;   // clamp A reads; stores are guarded
    int n = nt * 16 + r;
    float bz = bias ? bias[n] : 0.0f;
    v8f c;
    #pragma unroll
    for (int j = 0; j < 8; ++j) c[j] = bz;
    for (int k0 = 0; k0 < K; k0 += 4) {
        int ka = k0 + 2 * hi;
        v2f a, b;
        a.x = A[(size_t)m * K + ka];
        a.y = A[(size_t)m * K + ka + 1];
        b.x = B[(size_t)ka * Ncol + n];
        b.y = B[(size_t)(ka + 1) * Ncol + n];
        c = __builtin_amdgcn_wmma_f32_16x16x4_f32(false, a, false, b, (short)0, c,
                                                  false, false);
    }
    #pragma unroll
    for (int j = 0; j < 8; ++j) {
        int row = mt * 16 + j + 8 * hi;
        if (row < M) C[(size_t)row * Ncol + n] = c[j];
    }
}

// ---------- 5) per-row logit dots: out[r,h] = rows[r,:] . watt[:,h] ----------
__global__ void k_row_dots(const float* __restrict__ rows, const float* __restrict__ watt,
                           float* __restrict__ out, int nrows) {
    int tid = blockIdx.x * blockDim.x + threadIdx.x;
    if (tid >= nrows * 4) return;
    int r = tid >> 2, h = tid & 3;
    float s = 0.f;
    #pragma unroll 8
    for (int d = 0; d < 64; ++d) s += rows[(size_t)r * 64 + d] * watt[d * 4 + h];
    out[tid] = s;
}

// ---------- 6) edge logits + leaky_relu + segment max ----------
__global__ void k_alpha_max(const float* __restrict__ a_high, const float* __restrict__ a_node,
                            const int* __restrict__ src, const int* __restrict__ dst,
                            float* __restrict__ alpha_raw, unsigned* __restrict__ amax_u, int E) {
    int tid = blockIdx.x * blockDim.x + threadIdx.x;
    if (tid >= E * 4) return;
    int e = tid >> 2, h = tid & 3;
    float a = a_high[(size_t)dst[e] * 4 + h] + a_node[(size_t)src[e] * 4 + h];
    a = (a > 0.f) ? a : 0.2f * a;           // leaky_relu(0.2)
    alpha_raw[tid] = a;
    atomicMax(&amax_u[(size_t)dst[e] * 4 + h], enc_f(a));
}

// ---------- 7) exp(shifted) + segment sum ----------
__global__ void k_exp_sum(float* __restrict__ alpha_raw, const unsigned* __restrict__ amax_u,
                          const int* __restrict__ dst, float* __restrict__ exsum, int E) {
    int tid = blockIdx.x * blockDim.x + threadIdx.x;
    if (tid >= E * 4) return;
    int e = tid >> 2, h = tid & 3;
    float ex = expf(alpha_raw[tid] - dec_f(amax_u[(size_t)dst[e] * 4 + h]));
    alpha_raw[tid] = ex;
    atomicAdd(&exsum[(size_t)dst[e] * 4 + h], ex);
}

// ---------- 8) normalize alpha -> d_out ----------
__global__ void k_alpha_norm(const float* __restrict__ alpha_raw, const float* __restrict__ exsum,
                             const int* __restrict__ dst, float* __restrict__ alpha_out, int E) {
    int tid = blockIdx.x * blockDim.x + threadIdx.x;
    if (tid >= E * 4) return;
    int e = tid >> 2, h = tid & 3;
    alpha_out[tid] = alpha_raw[tid] / exsum[(size_t)dst[e] * 4 + h];
}

// ---------- 9) h = mean_h(alpha * X[src]) scattered into seg_h ----------
__global__ void k_edge_agg(const float* __restrict__ X, const float* __restrict__ alpha,
                           const int* __restrict__ src, const int* __restrict__ dst,
                           float* __restrict__ seg_h, int E) {
    int tid = blockIdx.x * blockDim.x + threadIdx.x;
    if (tid >= E * 64) return;
    int e = tid >> 6, d = tid & 63;
    int s = src[e], t = dst[e];
    float acc = 0.f;
    #pragma unroll
    for (int h = 0; h < 4; ++h)
        acc += X[(size_t)s * 256 + h * 64 + d] * alpha[(size_t)e * 4 + h];
    atomicAdd(&seg_h[(size_t)t * 64 + d], 0.25f * acc);
}

extern "C" void kernel_launch(void* const* d_in, const int* in_sizes, int n_in,
                              void* d_out, int out_size, void* d_ws, size_t ws_size,
                              hipStream_t stream) {
    const float* F        = (const float*)d_in[0];   // [N,64]
    const float* W_src    = (const float*)d_in[1];   // [64,256]
    const float* W_dst    = (const float*)d_in[2];   // [64,256]
    const float* att_src  = (const float*)d_in[3];   // [1,4,64]
    const float* att_dst  = (const float*)d_in[4];   // [1,4,64]
    const float* W_high   = (const float*)d_in[5];   // [64,64]
    const float* b_high   = (const float*)d_in[6];   // [64]
    const int*   src      = (const int*)d_in[7];     // [E]
    const int*   dst      = (const int*)d_in[8];     // [E]

    const int N = in_sizes[0] / 64;
    const int E = in_sizes[7];
    const int H = (out_size - 4 * E) / 64;           // out = [H*64 | E*4]

    // ---- workspace layout (floats) ----
    float* ws = (float*)d_ws;
    float*    X         = ws;                              // N*256
    float*    sumfeat   = X        + (size_t)N * 256;      // H*64   (zeroed) -> h_mean
    float*    seg_h     = sumfeat  + (size_t)H * 64;       // H*64   (zeroed)
    float*    exsum     = seg_h    + (size_t)H * 64;       // H*4    (zeroed)
    unsigned* amax_u    = (unsigned*)(exsum + (size_t)H * 4); // H*4 (zeroed)
    float*    cnt       = (float*)(amax_u + (size_t)H * 4);   // H   (zeroed)
    float*    alpha_raw = cnt      + (size_t)H;            // E*4
    float*    a_node    = alpha_raw+ (size_t)E * 4;        // N*4
    float*    a_high    = a_node   + (size_t)N * 4;        // H*4
    float*    w_src_att = a_high   + (size_t)H * 4;        // 256
    float*    w_dst_att = w_src_att + 256;                 // 256

    float* out_hf    = (float*)d_out;                      // [H,64]
    float* out_alpha = out_hf + (size_t)H * 64;            // [E,4]

    // zero all accumulator buffers in one contiguous async memset (graph-safe)
    size_t zero_floats = (size_t)H * 64 + (size_t)H * 64 + (size_t)H * 4
                       + (size_t)H * 4 + (size_t)H;        // = H*137
    hipMemsetAsync(sumfeat, 0, zero_floats * sizeof(float), stream);

    const int TB = 256;
    auto cdiv = [](long long a, long long b) { return (int)((a + b - 1) / b); };

    // 1) fold attention into weights
    k_fold_att<<<1, 256, 0, stream>>>(W_src, W_dst, att_src, att_dst, w_src_att, w_dst_att);

    // 2) scatter-mean numerator + counts
    k_scatter_mean<<<cdiv((long long)E * 64, TB), TB, 0, stream>>>(F, src, dst, sumfeat, cnt, E);

    // 3) h_mean in place
    k_mean_div<<<cdiv((long long)H * 64, TB), TB, 0, stream>>>(sumfeat, cnt, H);

    // 4) X = F @ W_src   (fp32 WMMA)
    {
        long long waves = (long long)((N + 15) / 16) * (256 / 16);
        k_gemm_wmma_f32<<<cdiv(waves * 32, TB), TB, 0, stream>>>(F, W_src, nullptr, X, N, 256, 64);
    }

    // 5) per-row logit dots
    k_row_dots<<<cdiv((long long)N * 4, TB), TB, 0, stream>>>(F, w_src_att, a_node, N);
    k_row_dots<<<cdiv((long long)H * 4, TB), TB, 0, stream>>>(sumfeat, w_dst_att, a_high, H);

    // 6-8) segment softmax over dst
    k_alpha_max<<<cdiv((long long)E * 4, TB), TB, 0, stream>>>(a_high, a_node, src, dst,
                                                              alpha_raw, amax_u, E);
    k_exp_sum<<<cdiv((long long)E * 4, TB), TB, 0, stream>>>(alpha_raw, amax_u, dst, exsum, E);
    k_alpha_norm<<<cdiv((long long)E * 4, TB), TB, 0, stream>>>(alpha_raw, exsum, dst, out_alpha, E);

    // 9) weighted head-mean gather/scatter
    k_edge_agg<<<cdiv((long long)E * 64, TB), TB, 0, stream>>>(X, out_alpha, src, dst, seg_h, E);

    // 10) h_features = seg_h @ W_high + b_high   (fp32 WMMA, bias in accumulator)
    {
        long long waves = (long long)((H + 15) / 16) * (64 / 16);
        k_gemm_wmma_f32<<<cdiv(waves * 32, TB), TB, 0, stream>>>(seg_h, W_high, b_high, out_hf,
                                                                 H, 64, 64);
    }
}